// GraphGATClassifier_87608742904543
// MI455X (gfx1250) — compile-verified
//
#include <hip/hip_runtime.h>
#include <math.h>

#define N_NODES  100000
#define N_EDGES  1600000
#define N_GRAPHS 256
#define HID      128
#define NCLS     3

typedef float v2f __attribute__((ext_vector_type(2)));
typedef float v8f __attribute__((ext_vector_type(8)));

// ---------------------------------------------------------------------------
// float atomic max via signed/unsigned integer ordering trick (init = -inf)
// ---------------------------------------------------------------------------
__device__ __forceinline__ void atomicMaxF(float* addr, float val) {
    if (val >= 0.0f)
        atomicMax((int*)addr, __float_as_int(val));
    else
        atomicMin((unsigned int*)addr, __float_as_uint(val));
}

// ---------------------------------------------------------------------------
// utility fill
// ---------------------------------------------------------------------------
__global__ void k_fill(float* __restrict__ p, float v, int n) {
    int i = blockIdx.x * 256 + threadIdx.x;
    if (i < n) p[i] = v;
}

// ---------------------------------------------------------------------------
// in-degree (segment_sum of ones over dst)
// ---------------------------------------------------------------------------
__global__ void k_indeg(const int* __restrict__ dst, float* __restrict__ indeg) {
    int t = blockIdx.x * 256 + threadIdx.x;
    if (t < N_EDGES) atomicAdd(&indeg[dst[t]], 1.0f);
}

// ---------------------------------------------------------------------------
// layer-1 scalars: cl = W1 . al1, cr = W1 . ar1   (W1 is 1 x HID)
// ---------------------------------------------------------------------------
__global__ void k_scal1(const float* __restrict__ W1, const float* __restrict__ al1,
                        const float* __restrict__ ar1, float* __restrict__ clcr) {
    float cl = 0.f, cr = 0.f;
    for (int j = 0; j < HID; ++j) { cl += W1[j] * al1[j]; cr += W1[j] * ar1[j]; }
    clcr[0] = cl; clcr[1] = cr;
}

// ---------------------------------------------------------------------------
// layer-1 z (rank-1) plus el/er
// ---------------------------------------------------------------------------
__global__ void k_layer1(const float* __restrict__ indeg, const float* __restrict__ W1,
                         const float* __restrict__ clcr, float* __restrict__ Z,
                         float* __restrict__ el, float* __restrict__ er) {
    int idx = blockIdx.x * 256 + threadIdx.x;
    if (idx >= N_NODES * HID) return;
    int i = idx >> 7, j = idx & 127;
    float h = indeg[i];
    Z[idx] = h * W1[j];
    if (j == 0) { el[i] = h * clcr[0]; er[i] = h * clcr[1]; }
}

// ---------------------------------------------------------------------------
// fp32 WMMA GEMM: Z[N_NODES x HID] = H[N_NODES x HID] @ W[HID x HID]
// one wave -> 16x16 C tile via 32x v_wmma_f32_16x16x4_f32; 8 waves cover N=128
// block handles 16 rows (N_NODES % 16 == 0, EXEC always full as WMMA requires)
// ---------------------------------------------------------------------------
__global__ void __launch_bounds__(256)
k_gemm(const float* __restrict__ H, const float* __restrict__ W, float* __restrict__ Z) {
    __shared__ float sh[16 * HID];
    const int row0 = blockIdx.x * 16;
    const int lane = threadIdx.x & 31;
    const int wave = threadIdx.x >> 5;

    for (int i = threadIdx.x; i < 16 * HID; i += 256)
        sh[i] = H[(size_t)row0 * HID + i];
    __syncthreads();

    const int n0   = wave * 16;
    const int mrow = lane & 15;
    const int kh   = (lane >> 4) << 1;   // lanes 16-31 hold K+2 (per ISA layout)

    v8f c = {0.f, 0.f, 0.f, 0.f, 0.f, 0.f, 0.f, 0.f};
    #pragma unroll 8
    for (int k = 0; k < HID; k += 4) {
        v2f a, b;
        a.x = sh[mrow * HID + k + kh];
        a.y = sh[mrow * HID + k + kh + 1];
        b.x = W[(size_t)(k + kh) * HID + n0 + mrow];
        b.y = W[(size_t)(k + kh + 1) * HID + n0 + mrow];
        c = __builtin_amdgcn_wmma_f32_16x16x4_f32(false, a, false, b,
                                                  (short)0, c, false, false);
    }

    const int rhi = (lane >> 4) << 3;    // D: row = v + 8*(lane/16), col = lane%16
    const int col = lane & 15;
    #pragma unroll
    for (int v = 0; v < 8; ++v)
        Z[(size_t)(row0 + rhi + v) * HID + n0 + col] = c[v];
}

// ---------------------------------------------------------------------------
// el/er = Z . al / Z . ar (per node)
// ---------------------------------------------------------------------------
__global__ void k_elr(const float* __restrict__ Z, const float* __restrict__ al,
                      const float* __restrict__ ar, float* __restrict__ el,
                      float* __restrict__ er) {
    int i = blockIdx.x * 256 + threadIdx.x;
    if (i >= N_NODES) return;
    const float4* zr = (const float4*)(Z + (size_t)i * HID);
    float cl = 0.f, cr = 0.f;
    #pragma unroll 8
    for (int j = 0; j < HID / 4; ++j) {
        float4 z4 = zr[j];
        float4 a4 = ((const float4*)al)[j];
        float4 r4 = ((const float4*)ar)[j];
        cl += z4.x * a4.x + z4.y * a4.y + z4.z * a4.z + z4.w * a4.w;
        cr += z4.x * r4.x + z4.y * r4.y + z4.z * r4.z + z4.w * r4.w;
    }
    el[i] = cl; er[i] = cr;
}

// ---------------------------------------------------------------------------
// edge pass 1: e = leaky_relu(el[src]+er[dst]); segment max into m[dst]
// ---------------------------------------------------------------------------
__global__ void k_edge_att(const float* __restrict__ el, const float* __restrict__ er,
                           const int* __restrict__ src, const int* __restrict__ dst,
                           float* __restrict__ eb, float* __restrict__ m) {
    int t = blockIdx.x * 256 + threadIdx.x;
    if (t >= N_EDGES) return;
    float x = el[src[t]] + er[dst[t]];
    x = (x > 0.f) ? x : 0.2f * x;
    eb[t] = x;
    atomicMaxF(&m[dst[t]], x);
}

// ---------------------------------------------------------------------------
// edge pass 2: ex = exp(e - m[dst]) (in place); segment sum into s[dst]
// ---------------------------------------------------------------------------
__global__ void k_edge_exp(const int* __restrict__ dst, const float* __restrict__ m,
                           float* __restrict__ eb, float* __restrict__ s) {
    int t = blockIdx.x * 256 + threadIdx.x;
    if (t >= N_EDGES) return;
    int d = dst[t];
    float v = expf(eb[t] - m[d]);
    eb[t] = v;
    atomicAdd(&s[d], v);
}

// ---------------------------------------------------------------------------
// edge scatter: acc[dst,:] += alpha * z[src,:]  (32 lanes/edge, float4 each)
// ---------------------------------------------------------------------------
__global__ void k_scatter(const float* __restrict__ Z, const int* __restrict__ src,
                          const int* __restrict__ dst, const float* __restrict__ ex,
                          const float* __restrict__ s, float* __restrict__ acc) {
    unsigned idx = blockIdx.x * 256u + threadIdx.x;
    unsigned e = idx >> 5;
    if (e >= N_EDGES) return;
    unsigned f = (idx & 31u) * 4u;
    int sn = src[e], dn = dst[e];
    float alpha = ex[e] / fmaxf(s[dn], 1e-9f);
    float4 zv = *(const float4*)(Z + (size_t)sn * HID + f);
    float* o = acc + (size_t)dn * HID + f;
    atomicAdd(o + 0, alpha * zv.x);
    atomicAdd(o + 1, alpha * zv.y);
    atomicAdd(o + 2, alpha * zv.z);
    atomicAdd(o + 3, alpha * zv.w);
}

// ---------------------------------------------------------------------------
// bias + ELU in place
// ---------------------------------------------------------------------------
__global__ void k_bias_elu(float* __restrict__ acc, const float* __restrict__ b) {
    int idx = blockIdx.x * 256 + threadIdx.x;
    if (idx >= N_NODES * HID) return;
    float x = acc[idx] + b[idx & 127];
    acc[idx] = (x > 0.f) ? x : (expf(x) - 1.f);
}

// ---------------------------------------------------------------------------
// graph mean-pool: sums[g,:] += h[i,:]; cnts[g] += 1
// ---------------------------------------------------------------------------
__global__ void k_pool(const float* __restrict__ h, const int* __restrict__ gid,
                       float* __restrict__ sums, float* __restrict__ cnts) {
    int idx = blockIdx.x * 256 + threadIdx.x;
    if (idx >= N_NODES * HID) return;
    int i = idx >> 7, j = idx & 127;
    int g = gid[i];
    atomicAdd(&sums[g * HID + j], h[idx]);
    if (j == 0) atomicAdd(&cnts[g], 1.0f);
}

// ---------------------------------------------------------------------------
// classifier: out[g,c] = (sums[g,:]/max(cnt,1)) . Wc[:,c] + bc[c]
// ---------------------------------------------------------------------------
__global__ void k_classify(const float* __restrict__ sums, const float* __restrict__ cnts,
                           const float* __restrict__ Wc, const float* __restrict__ bc,
                           float* __restrict__ out) {
    int idx = blockIdx.x * 256 + threadIdx.x;
    if (idx >= N_GRAPHS * NCLS) return;
    int g = idx / NCLS, c = idx % NCLS;
    float inv = 1.0f / fmaxf(cnts[g], 1.0f);
    float acc = 0.f;
    for (int j = 0; j < HID; ++j)
        acc += sums[g * HID + j] * Wc[j * NCLS + c];
    out[idx] = acc * inv + bc[c];
}

// ---------------------------------------------------------------------------
extern "C" void kernel_launch(void* const* d_in, const int* in_sizes, int n_in,
                              void* d_out, int out_size, void* d_ws, size_t ws_size,
                              hipStream_t stream) {
    const int*   edge_src  = (const int*)d_in[0];
    const int*   edge_dst  = (const int*)d_in[1];
    const int*   graph_ids = (const int*)d_in[2];
    const float* W1  = (const float*)d_in[3];
    const float* al1 = (const float*)d_in[4];
    const float* ar1 = (const float*)d_in[5];
    const float* b1  = (const float*)d_in[6];
    const float* W2  = (const float*)d_in[7];
    const float* al2 = (const float*)d_in[8];
    const float* ar2 = (const float*)d_in[9];
    const float* b2  = (const float*)d_in[10];
    const float* Wc  = (const float*)d_in[11];
    const float* bc  = (const float*)d_in[12];
    float* out = (float*)d_out;

    // workspace layout (floats); ~111 MB total
    float* A     = (float*)d_ws;                      // z buffer      N*HID
    float* B     = A + (size_t)N_NODES * HID;         // acc / h       N*HID
    float* el    = B + (size_t)N_NODES * HID;         // N
    float* er    = el + N_NODES;                      // N
    float* m     = er + N_NODES;                      // N
    float* s     = m + N_NODES;                       // N
    float* eb    = s + N_NODES;                       // E
    float* indeg = eb + N_EDGES;                      // N
    float* sums  = indeg + N_NODES;                   // G*HID
    float* cnts  = sums + (size_t)N_GRAPHS * HID;     // G
    float* clcr  = cnts + N_GRAPHS;                   // 2

    const int NB_N  = (N_NODES + 255) / 256;
    const int NB_E  = (N_EDGES + 255) / 256;
    const int NB_NH = (N_NODES * HID + 255) / 256;
    const int NB_SC = (N_EDGES * 32) / 256;           // exact: 200000 blocks

    // in-degree
    k_fill<<<NB_N, 256, 0, stream>>>(indeg, 0.f, N_NODES);
    k_indeg<<<NB_E, 256, 0, stream>>>(edge_dst, indeg);

    // layer-1 z (rank-1) and attention logits
    k_scal1<<<1, 1, 0, stream>>>(W1, al1, ar1, clcr);
    k_layer1<<<NB_NH, 256, 0, stream>>>(indeg, W1, clcr, A, el, er);

    for (int layer = 0; layer < 2; ++layer) {
        k_fill<<<NB_N, 256, 0, stream>>>(m, -__builtin_inff(), N_NODES);
        k_fill<<<NB_N, 256, 0, stream>>>(s, 0.f, N_NODES);
        k_fill<<<NB_NH, 256, 0, stream>>>(B, 0.f, N_NODES * HID);

        k_edge_att<<<NB_E, 256, 0, stream>>>(el, er, edge_src, edge_dst, eb, m);
        k_edge_exp<<<NB_E, 256, 0, stream>>>(edge_dst, m, eb, s);
        k_scatter<<<NB_SC, 256, 0, stream>>>(A, edge_src, edge_dst, eb, s, B);
        k_bias_elu<<<NB_NH, 256, 0, stream>>>(B, layer == 0 ? b1 : b2);

        if (layer == 0) {
            // layer-2 GEMM: z2 = h1 @ W2 via fp32 WMMA
            k_gemm<<<N_NODES / 16, 256, 0, stream>>>(B, W2, A);
            k_elr<<<NB_N, 256, 0, stream>>>(A, al2, ar2, el, er);
        }
    }

    // graph mean-pool + classifier
    k_fill<<<(N_GRAPHS * HID + 255) / 256, 256, 0, stream>>>(sums, 0.f, N_GRAPHS * HID);
    k_fill<<<1, 256, 0, stream>>>(cnts, 0.f, N_GRAPHS);
    k_pool<<<NB_NH, 256, 0, stream>>>(B, graph_ids, sums, cnts);
    k_classify<<<(N_GRAPHS * NCLS + 255) / 256, 256, 0, stream>>>(sums, cnts, Wc, bc, out);
}